// soft_cldice_loss_20358144983652
// MI455X (gfx1250) — compile-verified
//
#include <hip/hip_runtime.h>

// ---------------- problem geometry ----------------
#define Dd 128
#define Hd 256
#define Wd 256
#define NBATCH 2

#define THt 8    // H tile (interior)
#define TWt 32   // W tile (interior)
#define ZCH 32   // z-planes marched per block

#define NZB (Dd / ZCH)                       // 4
#define NB_W (Wd / TWt)                      // 8
#define NB_H (Hd / THt)                      // 32
#define BLOCKS_PER_BATCH (NB_W * NB_H * NZB) // 1024
#define NBLK (BLOCKS_PER_BATCH * NBATCH)     // 2048
#define NTHREADS 256

#define PXp 36                 // padded plane width  (TWt + 4)
#define PYp 12                 // padded plane height (THt + 4)
#define PPLANE (PXp * PYp)     // 432 floats per plane per tensor
#define LOADTOT (2 * PPLANE)   // 864 (both tensors)
#define EPOS 340               // 10*34 eroded positions per plane

// ---------------- CDNA5 async global->LDS path ----------------
#if defined(__HIP_DEVICE_COMPILE__) && __has_builtin(__builtin_amdgcn_global_load_async_to_lds_b32)
#define USE_ASYNC_LDS 1
#else
#define USE_ASYNC_LDS 0
#endif

typedef __attribute__((address_space(1))) int* gint_ptr;
typedef __attribute__((address_space(3))) int* lint_ptr;

__device__ __forceinline__ void async_g2l_b32(const float* g, float* l) {
#if USE_ASYNC_LDS
  __builtin_amdgcn_global_load_async_to_lds_b32((gint_ptr)(g), (lint_ptr)(l),
                                                /*imm offset*/ 0, /*cpol*/ 0);
#else
  *l = *g;  // fallback: plain load + ds_store
#endif
}

// wait until at most 4 async-LDS ops remain outstanding (the prefetched plane)
__device__ __forceinline__ void wait_async_le4() {
#if defined(__HIP_DEVICE_COMPILE__)
#if __has_builtin(__builtin_amdgcn_s_wait_asynccnt)
  __builtin_amdgcn_s_wait_asynccnt(4);
#else
  asm volatile("s_wait_asynccnt 4" ::: "memory");
#endif
#endif
}

__device__ __forceinline__ int iclamp(int v, int lo, int hi) {
  return v < lo ? lo : (v > hi ? hi : v);
}

// 3x3 min / max over an LDS plane (row-major, pitch PXp); imm-offset ds loads
__device__ __forceinline__ float min9(const float* p, int o) {
  float a = fminf(fminf(p[o - PXp - 1], p[o - PXp]), p[o - PXp + 1]);
  float b = fminf(fminf(p[o - 1], p[o]), p[o + 1]);
  float c = fminf(fminf(p[o + PXp - 1], p[o + PXp]), p[o + PXp + 1]);
  return fminf(fminf(a, b), c);
}
__device__ __forceinline__ float max9(const float* p, int o) {
  float a = fmaxf(fmaxf(p[o - PXp - 1], p[o - PXp]), p[o - PXp + 1]);
  float b = fmaxf(fmaxf(p[o - 1], p[o]), p[o + 1]);
  float c = fmaxf(fmaxf(p[o + PXp - 1], p[o + PXp]), p[o + PXp + 1]);
  return fmaxf(fmaxf(a, b), c);
}

// ---------------- main fused kernel ----------------
// Per block: one (8 x 32) H,W column, marching ZCH z-planes.
// z-direction of erosion (min) and dilation (max) lives in register rings;
// only the 2-D 3x3 windows touch LDS. Planes are double-buffered and filled by
// async global->LDS DMA, synchronized with s_wait_asynccnt.
// partials: p[0]=sum(skelP*yt) p[1]=sum(skelP) p[2]=sum(skelT*yp) p[3]=sum(skelT)
__global__ __launch_bounds__(NTHREADS) void cldice_main(
    const float* __restrict__ yp, const float* __restrict__ yt,
    float* __restrict__ partials) {
  __shared__ float xs[2][2][PPLANE];  // [tensor][parity][plane]
  __shared__ float es[2][PPLANE];     // eroded plane (reused every step)
  __shared__ float sWave[NTHREADS / 32][4];

  const int tid = threadIdx.x;
  const int blk = blockIdx.x;
  const int wb = blk & 7;
  const int hb = (blk >> 3) & 31;
  const int zb = (blk >> 8) & 3;
  const int b = blk >> 10;

  const int w0 = wb * TWt, h0 = hb * THt;
  const int z0 = zb * ZCH, z1 = z0 + ZCH;

  const float* ypb = yp + (size_t)b * Dd * Hd * Wd;
  const float* ytb = yt + (size_t)b * Dd * Hd * Wd;

  // ---- per-thread precompute: load slots (4 per thread, uniform per wave) ----
  int lt[4], lpi[4], lyx[4];
#pragma unroll
  for (int k = 0; k < 4; ++k) {
    int li = tid + k * NTHREADS;
    if (li > LOADTOT - 1) li = LOADTOT - 1;  // duplicate tail -> uniform count
    const int t = li >= PPLANE;
    const int pi = li - t * PPLANE;
    const int py = pi / PXp;
    const int px = pi - py * PXp;
    const int gy = iclamp(h0 + py - 2, 0, Hd - 1);  // edge replication
    const int gx = iclamp(w0 + px - 2, 0, Wd - 1);
    lt[k] = t;
    lpi[k] = pi;
    lyx[k] = gy * Wd + gx;
  }

  // ---- per-thread precompute: eroded-plane slots (<=2) ----
  const int e0 = tid;             // always < EPOS
  const int e1 = NTHREADS + tid;  // valid iff tid < EPOS-NTHREADS (=84)
  const bool has1 = (e1 < EPOS);
  const int ey0 = 1 + e0 / 34, ex0 = 1 + (e0 - (e0 / 34) * 34);
  const int ey1 = 1 + e1 / 34, ex1 = 1 + (e1 - (e1 / 34) * 34);
  const int el0 = ey0 * PXp + ex0;
  const int el1 = ey1 * PXp + ex1;
  const bool oob0 = ((unsigned)(h0 + ey0 - 2) >= (unsigned)Hd) ||
                    ((unsigned)(w0 + ex0 - 2) >= (unsigned)Wd);
  const bool oob1 = ((unsigned)(h0 + ey1 - 2) >= (unsigned)Hd) ||
                    ((unsigned)(w0 + ex1 - 2) >= (unsigned)Wd);

  // interior position (1 per thread), always globally valid
  const int il = (2 + (tid >> 5)) * PXp + (2 + (tid & 31));

  const float NINF = -__builtin_inff();

  // register rings: [tensor][slot][age]  age: 0 = oldest
  float m2r[2][2][3], n2r[2][3], xcr[2][3];
#pragma unroll
  for (int t = 0; t < 2; ++t) {
#pragma unroll
    for (int r = 0; r < 3; ++r) {
      m2r[t][0][r] = 0.f;
      m2r[t][1][r] = 0.f;
      n2r[t][r] = NINF;
      xcr[t][r] = 0.f;
    }
  }

  float s0 = 0.f, s1 = 0.f, s2 = 0.f, s3 = 0.f;

  // issue-loads helper (exactly 4 async instructions per wave)
  auto issue_plane = [&](int pz) {
    const int gz = iclamp(pz, 0, Dd - 1);  // z edge replication
    const int par = pz & 1;
    const size_t zoff = (size_t)gz * (Hd * Wd);
#pragma unroll
    for (int k = 0; k < 4; ++k) {
      const float* g = (lt[k] ? ytb : ypb) + zoff + lyx[k];
      async_g2l_b32(g, &xs[lt[k]][par][lpi[k]]);
    }
  };

  // ---- pipeline: prologue load, then march ----
  issue_plane(z0 - 2);

  for (int pz = z0 - 2; pz <= z1 + 1; ++pz) {
    issue_plane(pz + 1);  // prefetch next plane into other parity
    wait_async_le4();     // plane pz has landed
    __syncthreads();

    const int par = pz & 1;

    // 2-D 3x3 min (m2) of plane pz + capture interior center value
#pragma unroll
    for (int t = 0; t < 2; ++t) {
      const float* xp = xs[t][par];
      m2r[t][0][2] = min9(xp, el0);
      if (has1) m2r[t][1][2] = min9(xp, el1);
      xcr[t][2] = xp[il];
    }

    // eroded plane ze = pz-1 : z-min in registers, -inf at global-OOB
    const int ze = pz - 1;
    const bool zin = (ze >= 0) && (ze < Dd);
#pragma unroll
    for (int t = 0; t < 2; ++t) {
      float er = fminf(fminf(m2r[t][0][0], m2r[t][0][1]), m2r[t][0][2]);
      es[t][el0] = (zin && !oob0) ? er : NINF;
      if (has1) {
        float er1v = fminf(fminf(m2r[t][1][0], m2r[t][1][1]), m2r[t][1][2]);
        es[t][el1] = (zin && !oob1) ? er1v : NINF;
      }
    }
    __syncthreads();

    // 2-D 3x3 max (n2) of eroded plane ze at interior position
#pragma unroll
    for (int t = 0; t < 2; ++t) n2r[t][2] = max9(es[t], il);

    // opened plane zo = pz-2 : z-max in registers; skeleton + accumulate
    const int zo = pz - 2;
    if (zo >= z0 && zo < z1) {
      const float opP = fmaxf(fmaxf(n2r[0][0], n2r[0][1]), n2r[0][2]);
      const float opT = fmaxf(fmaxf(n2r[1][0], n2r[1][1]), n2r[1][2]);
      const float xP = xcr[0][0], xT = xcr[1][0];
      const float skP = fmaxf(xP - opP, 0.0f);
      const float skT = fmaxf(xT - opT, 0.0f);
      s0 += skP * xT;
      s1 += skP;
      s2 += skT * xP;
      s3 += skT;
    }

    // rotate rings
#pragma unroll
    for (int t = 0; t < 2; ++t) {
      m2r[t][0][0] = m2r[t][0][1];
      m2r[t][0][1] = m2r[t][0][2];
      m2r[t][1][0] = m2r[t][1][1];
      m2r[t][1][1] = m2r[t][1][2];
      n2r[t][0] = n2r[t][1];
      n2r[t][1] = n2r[t][2];
      xcr[t][0] = xcr[t][1];
      xcr[t][1] = xcr[t][2];
    }
  }

  // ---- deterministic block reduction (wave32 shuffles -> LDS -> serial) ----
#pragma unroll
  for (int off = 16; off > 0; off >>= 1) {
    s0 += __shfl_down(s0, off, 32);
    s1 += __shfl_down(s1, off, 32);
    s2 += __shfl_down(s2, off, 32);
    s3 += __shfl_down(s3, off, 32);
  }
  const int wave = tid >> 5, lane = tid & 31;
  if (lane == 0) {
    sWave[wave][0] = s0;
    sWave[wave][1] = s1;
    sWave[wave][2] = s2;
    sWave[wave][3] = s3;
  }
  __syncthreads();
  if (tid == 0) {
    float a0 = 0.f, a1 = 0.f, a2 = 0.f, a3 = 0.f;
    for (int w = 0; w < NTHREADS / 32; ++w) {
      a0 += sWave[w][0];
      a1 += sWave[w][1];
      a2 += sWave[w][2];
      a3 += sWave[w][3];
    }
    float* p = partials + (size_t)blk * 4;
    p[0] = a0;
    p[1] = a1;
    p[2] = a2;
    p[3] = a3;
  }
}

// ---------------- finalize: deterministic global reduction + loss ----------
__global__ __launch_bounds__(NTHREADS) void cldice_finalize(
    const float* __restrict__ partials, float* __restrict__ out) {
  __shared__ float red[NTHREADS][8];
  float acc[8];
#pragma unroll
  for (int j = 0; j < 8; ++j) acc[j] = 0.f;

  for (int blk = threadIdx.x; blk < NBLK; blk += NTHREADS) {
    const int b = blk / BLOCKS_PER_BATCH;
    const float* p = partials + (size_t)blk * 4;
#pragma unroll
    for (int j = 0; j < 4; ++j) acc[b * 4 + j] += p[j];
  }
#pragma unroll
  for (int j = 0; j < 8; ++j) red[threadIdx.x][j] = acc[j];
  __syncthreads();

  for (int s = NTHREADS / 2; s > 0; s >>= 1) {
    if (threadIdx.x < (unsigned)s) {
#pragma unroll
      for (int j = 0; j < 8; ++j) red[threadIdx.x][j] += red[threadIdx.x + s][j];
    }
    __syncthreads();
  }

  if (threadIdx.x == 0) {
    const float sm = 1e-5f;
    float csum = 0.f;
#pragma unroll
    for (int b = 0; b < NBATCH; ++b) {
      const float tprec = red[0][b * 4 + 0] / (red[0][b * 4 + 1] + sm);
      const float tsens = red[0][b * 4 + 2] / (red[0][b * 4 + 3] + sm);
      csum += 2.0f * (tprec * tsens) / (tprec + tsens + sm);
    }
    out[0] = 1.0f - csum / (float)NBATCH;
  }
}

// ---------------- launch ----------------
extern "C" void kernel_launch(void* const* d_in, const int* in_sizes, int n_in,
                              void* d_out, int out_size, void* d_ws,
                              size_t ws_size, hipStream_t stream) {
  (void)in_sizes;
  (void)n_in;
  (void)out_size;
  (void)ws_size;
  const float* yp = (const float*)d_in[0];
  const float* yt = (const float*)d_in[1];
  float* partials = (float*)d_ws;  // NBLK*4 floats = 32 KiB

  cldice_main<<<NBLK, NTHREADS, 0, stream>>>(yp, yt, partials);
  cldice_finalize<<<1, NTHREADS, 0, stream>>>(partials, (float*)d_out);
}